// DistortionLoss_35261681500899
// MI455X (gfx1250) — compile-verified
//
#include <hip/hip_runtime.h>

// ---------------------------------------------------------------------------
// Distortion loss: one wave32 per ray, segmented scan via wave shuffles,
// deterministic two-stage reduction (no atomics).
// CDNA5 path: double-buffered async global->LDS staging (ASYNCcnt),
// fallback: direct coalesced loads + global_prefetch_b8.
// ---------------------------------------------------------------------------

#define WPB            8               // waves (rays) per block
#define BLOCK_THREADS  (WPB * 32)
#define CHUNK          32

#if defined(__HIP_DEVICE_COMPILE__) && \
    __has_builtin(__builtin_amdgcn_global_load_async_to_lds_b32) && \
    __has_builtin(__builtin_amdgcn_s_wait_asynccnt)
#define USE_ASYNC 1
#else
#define USE_ASYNC 0
#endif

#if USE_ASYNC
typedef __attribute__((address_space(1))) int GInt;   // global int
typedef __attribute__((address_space(3))) int LInt;   // LDS int

__device__ __forceinline__ void async_g2l_b32(const float* g, float* l) {
  // global_load_async_to_lds_b32: per-lane 4-byte copy global -> LDS (ASYNCcnt)
  __builtin_amdgcn_global_load_async_to_lds_b32((GInt*)(const void*)g,
                                                (LInt*)(void*)l,
                                                /*imm offset*/0, /*cpol*/0);
}
#endif

__device__ __forceinline__ float incl_scan32(float x, int lane) {
#pragma unroll
  for (int off = 1; off < 32; off <<= 1) {
    float y = __shfl_up(x, off, 32);
    x += (lane >= off) ? y : 0.0f;
  }
  return x;
}

__device__ __forceinline__ void accum_chunk(float wv, float tv, float dv, int lane,
                                            float& carryW, float& carryWT, float& acc) {
  const float wt  = wv * tv;
  const float sw  = incl_scan32(wv, lane);   // inclusive prefix of ws
  const float swt = incl_scan32(wt, lane);   // inclusive prefix of ws*ts
  const float Wp  = carryW  + (sw  - wv);    // exclusive
  const float WTp = carryWT + (swt - wt);
  acc += 2.0f * wv * (tv * Wp - WTp) + wv * wv * dv * (1.0f / 3.0f);
  carryW  += __shfl(sw,  31, 32);
  carryWT += __shfl(swt, 31, 32);
}

__global__ __launch_bounds__(BLOCK_THREADS)
void distortion_stage1(const float* __restrict__ ws, const float* __restrict__ deltas,
                       const float* __restrict__ ts, const int* __restrict__ rays,
                       float* __restrict__ block_partial, int R) {
  __shared__ float wsum[WPB];
#if USE_ASYNC
  __shared__ float stage[WPB][2][3][CHUNK];  // [wave][buf][ws|ts|delta][lane]
#endif
  const int lane = threadIdx.x & 31;
  const int w    = threadIdx.x >> 5;
  const int ray  = blockIdx.x * WPB + w;

  float acc = 0.0f;
  if (ray < R) {
    const int start = rays[ray * 3 + 1];
    const int count = rays[ray * 3 + 2];
    const int end   = start + count;
    float carryW = 0.0f, carryWT = 0.0f;

#if USE_ASYNC
    if (start < end) {  // prologue: fill buffer 0
      const int i = start + lane;
      if (i < end) {
        async_g2l_b32(ws     + i, &stage[w][0][0][lane]);
        async_g2l_b32(ts     + i, &stage[w][0][1][lane]);
        async_g2l_b32(deltas + i, &stage[w][0][2][lane]);
      }
    }
    int buf = 0;
    for (int base = start; base < end; base += CHUNK) {
      const int nb = base + CHUNK;
      if (nb < end) {  // issue next chunk into the other buffer, then drain to <=3
        const int i = nb + lane;
        if (i < end) {
          async_g2l_b32(ws     + i, &stage[w][buf ^ 1][0][lane]);
          async_g2l_b32(ts     + i, &stage[w][buf ^ 1][1][lane]);
          async_g2l_b32(deltas + i, &stage[w][buf ^ 1][2][lane]);
        }
        __builtin_amdgcn_s_wait_asynccnt(3);
      } else {
        __builtin_amdgcn_s_wait_asynccnt(0);
      }
      const bool act = (base + lane) < end;
      const float wv = act ? stage[w][buf][0][lane] : 0.0f;
      const float tv = act ? stage[w][buf][1][lane] : 0.0f;
      const float dv = act ? stage[w][buf][2][lane] : 0.0f;
      accum_chunk(wv, tv, dv, lane, carryW, carryWT, acc);
      buf ^= 1;
    }
#else
    for (int base = start; base < end; base += CHUNK) {
      const int i   = base + lane;
      const bool act = i < end;
      if (i + 4 * CHUNK < end) {  // gfx1250: global_prefetch_b8 (speculative, safe)
        __builtin_prefetch(ws + i + 4 * CHUNK, 0, 3);
        __builtin_prefetch(ts + i + 4 * CHUNK, 0, 3);
        __builtin_prefetch(deltas + i + 4 * CHUNK, 0, 3);
      }
      const float wv = act ? ws[i]     : 0.0f;
      const float tv = act ? ts[i]     : 0.0f;
      const float dv = act ? deltas[i] : 0.0f;
      accum_chunk(wv, tv, dv, lane, carryW, carryWT, acc);
    }
#endif
  }

  // lane reduce within the wave (wave32)
#pragma unroll
  for (int off = 16; off > 0; off >>= 1) acc += __shfl_xor(acc, off, 32);
  if (lane == 0) wsum[w] = acc;
  __syncthreads();

  if (threadIdx.x == 0) {  // deterministic fixed-order block sum
    float b = 0.0f;
#pragma unroll
    for (int k = 0; k < WPB; ++k) b += wsum[k];
    block_partial[blockIdx.x] = b;
  }
}

__global__ __launch_bounds__(1024)
void distortion_stage2(const float* __restrict__ part, int n, float invR,
                       float* __restrict__ out) {
  __shared__ float s[1024];
  float a = 0.0f;
  for (int i = threadIdx.x; i < n; i += 1024) a += part[i];  // fixed stride order
  s[threadIdx.x] = a;
  __syncthreads();
  for (int st = 512; st > 0; st >>= 1) {
    if ((int)threadIdx.x < st) s[threadIdx.x] += s[threadIdx.x + st];
    __syncthreads();
  }
  if (threadIdx.x == 0) out[0] = s[0] * invR;
}

extern "C" void kernel_launch(void* const* d_in, const int* in_sizes, int n_in,
                              void* d_out, int out_size, void* d_ws, size_t ws_size,
                              hipStream_t stream) {
  (void)n_in; (void)out_size; (void)ws_size;
  const float* ws     = (const float*)d_in[0];
  const float* deltas = (const float*)d_in[1];
  const float* ts     = (const float*)d_in[2];
  const int*   rays   = (const int*)d_in[3];
  const int R = in_sizes[3] / 3;

  float* part = (float*)d_ws;                    // one partial per block (~32 KB)
  const int blocks = (R + WPB - 1) / WPB;

  distortion_stage1<<<blocks, BLOCK_THREADS, 0, stream>>>(ws, deltas, ts, rays, part, R);
  distortion_stage2<<<1, 1024, 0, stream>>>(part, blocks, 1.0f / (float)R,
                                            (float*)d_out);
}